// YOLOLoss_12249246729029
// MI455X (gfx1250) — compile-verified
//
#include <hip/hip_runtime.h>
#include <hip/hip_bf16.h>

// ---------------------------------------------------------------------------
// YOLO loss, restructured for MI455X:
//  - dense read limited to the objectness channel (~1 MB instead of 91 MB)
//  - sparse gathers at <=50 object cells per image per scale
//  - wave reduction via V_WMMA_F32_16X16X4_F32 (fp32-exact, 1 instr)
// Workspace layout (floats):
//   ws[  0.. 95] : dense sum of 1.5*softplus(po) per (scale,image)
//   ws[ 96..191] : obj correction at object cells
//   ws[192..287] : bbox squared-error sum
//   ws[288..383] : cls BCE sum
//   ws[384..479] : n_obj (unique object cells)
// ---------------------------------------------------------------------------

typedef float v2f __attribute__((ext_vector_type(2)));
typedef float v8f __attribute__((ext_vector_type(8)));

#define NUM_CLASSES 80
#define NCH 85
#define BATCH 32

__device__ __forceinline__ float softplusf(float x) {
    // log(1+exp(x)) = max(x,0) + log1p(exp(-|x|)), numerically stable
    return fmaxf(x, 0.0f) + log1pf(expf(-fabsf(x)));
}

__global__ void yolo_init_ws(float* __restrict__ ws) {
    int i = blockIdx.x * blockDim.x + threadIdx.x;
    if (i < 480) ws[i] = 0.0f;
}

// Dense pass: 268,800 elements = 1050 blocks * 256 threads exactly.
// Each half-wave lies entirely inside one (scale,image) chunk because every chunk
// size (6400/1600/400) and every chunk offset is a multiple of 16; a scale-2 wave can span two images.
__global__ __launch_bounds__(256) void yolo_dense_obj(
    const float* __restrict__ p3, const float* __restrict__ p4,
    const float* __restrict__ p5, float* __restrict__ ws) {
    int i = blockIdx.x * 256 + threadIdx.x;

    int s, b, off, HW;
    const float* base;
    if (i < 204800) {            // scale 0: 32 * 6400
        s = 0; HW = 6400; b = i / 6400; off = i % 6400; base = p3;
    } else if (i < 256000) {     // scale 1: 32 * 1600
        s = 1; HW = 1600; int r = i - 204800; b = r / 1600; off = r % 1600; base = p4;
    } else {                     // scale 2: 32 * 400
        s = 2; HW = 400;  int r = i - 256000; b = r / 400;  off = r % 400;  base = p5;
    }
    int sb = s * BATCH + b;

    float x = base[((size_t)b * NCH + 4) * (size_t)HW + off];
    float partial = 1.5f * softplusf(x);   // non-object contribution (1 + 0.5*noobj)

    int lane = threadIdx.x & 31;

#if defined(__has_builtin) && __has_builtin(__builtin_amdgcn_wmma_f32_16x16x4_f32)
    // A (16x4 f32): lanes 0-15 hold K=0..1, lanes 16-31 hold K=2..3 for M=lane%16.
    // a = {partial, 0} => rowsum_m = p_m + p_{m+16}.  B = all-ones (4x16) makes
    // D[m][n] = rowsum_m.  D layout: lanes 0-15 VGPR r -> M=r, lanes 16-31 -> M=r+8.
    // So sum(c[0..7]) on lane 0 = sum of partials {0..7,16..23}; on lane 16 the rest.
    v2f a;  a.x = partial; a.y = 0.0f;
    v2f bo; bo.x = 1.0f;   bo.y = 1.0f;
    v8f c = {};
    c = __builtin_amdgcn_wmma_f32_16x16x4_f32(false, a, false, bo,
                                              (short)0, c, false, false);
    float local = ((c[0] + c[1]) + (c[2] + c[3])) + ((c[4] + c[5]) + (c[6] + c[7]));
    if (lane == 0 || lane == 16) atomicAdd(&ws[sb], local);
#else
    float local = partial;
    for (int o = 16; o > 0; o >>= 1) local += __shfl_xor(local, o, 32);
    if (lane == 0) atomicAdd(&ws[sb], local);
#endif
}

// Sparse pass: one block per (scale,image). Dedup box->cell with
// last-writer-wins, then gather po/pb/pc at winning cells.
__global__ __launch_bounds__(128) void yolo_sparse(
    const float* __restrict__ p3, const float* __restrict__ p4,
    const float* __restrict__ p5, const float* __restrict__ tb,
    const int* __restrict__ tcls, float* __restrict__ ws) {
    int sb = blockIdx.x;
    int s = sb >> 5;           // /32
    int b = sb & 31;
    const float* base = (s == 0) ? p3 : ((s == 1) ? p4 : p5);
    int W  = (s == 0) ? 80 : ((s == 1) ? 40 : 20);
    int HW = W * W;

    __shared__ int   cellid[50];
    __shared__ int   win[50];
    __shared__ float bxs[50], bys[50], bws[50], bhs[50];
    __shared__ int   bcl[50];
    __shared__ float acc[4];   // corr, bbox, cls, nobj

    int t = threadIdx.x;
    if (t < 4) acc[t] = 0.0f;
    if (t < 50) {
        const float* p = tb + ((size_t)b * 50 + t) * 4;
        float x = p[0], y = p[1];
        bxs[t] = x; bys[t] = y; bws[t] = p[2]; bhs[t] = p[3];
        bcl[t] = tcls[b * 50 + t];
        int gx = min(max((int)(x * (float)W), 0), W - 1);
        int gy = min(max((int)(y * (float)W), 0), W - 1);
        cellid[t] = gy * W + gx;
    }
    __syncthreads();
    if (t < 50) {
        int wf = 1;
        for (int j = t + 1; j < 50; ++j)
            if (cellid[j] == cellid[t]) { wf = 0; break; }
        win[t] = wf;
    }
    __syncthreads();

    float corr = 0.0f, bsum = 0.0f, csum = 0.0f, ncnt = 0.0f;

    // obj correction + n_obj + bbox (per winning box)
    for (int j = t; j < 50; j += blockDim.x) {
        if (win[j]) {
            ncnt += 1.0f;
            int c = cellid[j];
            size_t cb = (size_t)b * NCH;
            float po = base[(cb + 4) * (size_t)HW + c];
            corr += -0.5f * softplusf(po) - po;   // (sp - po) - 1.5*sp
            float tg[4] = {bxs[j], bys[j], bws[j], bhs[j]};
#pragma unroll
            for (int k = 0; k < 4; ++k) {
                float pv = base[(cb + k) * (size_t)HW + c];
                float d = pv - tg[k];
                bsum += d * d;
            }
        }
    }

    // class BCE: 50 boxes x 80 classes
    for (int idx = t; idx < 50 * NUM_CLASSES; idx += blockDim.x) {
        int j = idx / NUM_CLASSES;
        int cc = idx - j * NUM_CLASSES;
        if (win[j]) {
            int c = cellid[j];
            float lg = base[((size_t)b * NCH + 5 + cc) * (size_t)HW + c];
            float tt = (cc == bcl[j]) ? 1.0f : 0.0f;
            csum += softplusf(lg) - lg * tt;
        }
    }

    atomicAdd(&acc[0], corr);
    atomicAdd(&acc[1], bsum);
    atomicAdd(&acc[2], csum);
    atomicAdd(&acc[3], ncnt);
    __syncthreads();
    if (t == 0) {
        ws[ 96 + sb] = acc[0];
        ws[192 + sb] = acc[1];
        ws[288 + sb] = acc[2];
        ws[384 + sb] = acc[3];
    }
}

__global__ __launch_bounds__(128) void yolo_finalize(
    const float* __restrict__ ws, float* __restrict__ out) {
    __shared__ float a[3];     // bbox, obj, cls
    int t = threadIdx.x;
    if (t < 3) a[t] = 0.0f;
    __syncthreads();
    if (t < 96) {
        int s = t >> 5;
        float HW = (s == 0) ? 6400.0f : ((s == 1) ? 1600.0f : 400.0f);
        float objL  = (ws[t] + ws[96 + t]) / HW;
        float bboxL = 5.0f * ws[192 + t];
        float nobj  = ws[384 + t];
        float clsL  = (nobj > 0.0f)
                        ? ws[288 + t] / fmaxf(nobj * (float)NUM_CLASSES, 1.0f)
                        : 0.0f;
        atomicAdd(&a[0], bboxL);
        atomicAdd(&a[1], objL);
        atomicAdd(&a[2], clsL);
    }
    __syncthreads();
    if (t == 0) {
        const float inv = 1.0f / 96.0f;   // B * 3 scales
        float bbox = a[0] * inv, obj = a[1] * inv, cls = a[2] * inv;
        out[0] = bbox;
        out[1] = obj;
        out[2] = cls;
        out[3] = bbox + obj + cls;
    }
}

extern "C" void kernel_launch(void* const* d_in, const int* in_sizes, int n_in,
                              void* d_out, int out_size, void* d_ws, size_t ws_size,
                              hipStream_t stream) {
    const float* p3 = (const float*)d_in[0];
    const float* p4 = (const float*)d_in[1];
    const float* p5 = (const float*)d_in[2];
    const float* tb = (const float*)d_in[3];
    const int*   tc = (const int*)d_in[4];
    float* ws  = (float*)d_ws;
    float* out = (float*)d_out;

    yolo_init_ws<<<4, 128, 0, stream>>>(ws);
    // 32*(6400+1600+400) = 268800 = 1050 * 256 exactly (full blocks -> EXEC all-ones)
    yolo_dense_obj<<<1050, 256, 0, stream>>>(p3, p4, p5, ws);
    yolo_sparse<<<96, 128, 0, stream>>>(p3, p4, p5, tb, tc, ws);
    yolo_finalize<<<1, 128, 0, stream>>>(ws, out);
}